// GATLayer_49297634623903
// MI455X (gfx1250) — compile-verified
//
#include <hip/hip_runtime.h>
#include <hip/hip_bf16.h>
#include <math.h>

#define GAT_N 10000
#define GAT_E 640000
#define GAT_FIN 128
#define GAT_NH 8
#define GAT_FOUT 16
#define GAT_NEG_SLOPE 0.2f
#define GAT_EPS 1e-16f

typedef __attribute__((ext_vector_type(2))) float v2f;
typedef __attribute__((ext_vector_type(8))) float v8f;

// ---------------------------------------------------------------------------
// Kernel 1: dual GEMM via V_WMMA_F32_16X16X4_F32.
//   proj = x @ W_proj^T   -> workspace
//   out  = x @ W_skip^T + bias   (initializes d_out; aggregation adds on top)
// One block per 16-row stripe of x (N = 625*16 exactly -> no remainder, EXEC
// all-ones as WMMA requires). 8 waves; wave w owns cols [16w, 16w+16) (== head
// w) and runs two accumulator chains sharing the A fragment.
// A fragment (16x4 f32, 2 VGPRs/lane): lanes 0-15 hold K={k0,k0+1} of row M=lane,
// lanes 16-31 hold K={k0+2,k0+3} of row M=lane-16.  B (4x16) mirrors this with
// N striped across lanes.  C/D: VGPR v holds row v (lanes 0-15) / v+8 (16-31).
// ---------------------------------------------------------------------------
__global__ __launch_bounds__(256) void gat_gemm_kernel(
    const float* __restrict__ x, const float* __restrict__ Wp,
    const float* __restrict__ Ws, const float* __restrict__ bias,
    float* __restrict__ proj, float* __restrict__ out)
{
    const int lane = threadIdx.x & 31;
    const int wave = threadIdx.x >> 5;   // 0..7 -> 16-col tile == head
    const int mb   = blockIdx.x << 4;    // row base
    const int half = lane >> 4;          // 0 or 1 (K sub-pair select)
    const int l16  = lane & 15;
    const int colb = wave << 4;

    const float* xp  = x  + (mb   + l16) * GAT_FIN + (half << 1);
    const float* wpp = Wp + (colb + l16) * GAT_FIN + (half << 1);
    const float* wsp = Ws + (colb + l16) * GAT_FIN + (half << 1);

    v8f accp = {0.f,0.f,0.f,0.f,0.f,0.f,0.f,0.f};
    v8f accs = {0.f,0.f,0.f,0.f,0.f,0.f,0.f,0.f};

#pragma unroll 4
    for (int k0 = 0; k0 < GAT_FIN; k0 += 4) {
        v2f a  = *(const v2f*)(xp  + k0);
        v2f bp = *(const v2f*)(wpp + k0);
        v2f bs = *(const v2f*)(wsp + k0);
        accp = __builtin_amdgcn_wmma_f32_16x16x4_f32(
            false, a, false, bp, (short)0, accp, false, false);
        accs = __builtin_amdgcn_wmma_f32_16x16x4_f32(
            false, a, false, bs, (short)0, accs, false, false);
    }

    const int   col  = colb + l16;
    const float bcol = bias[col];
#pragma unroll
    for (int v = 0; v < 8; ++v) {
        const int row = mb + v + (half << 3);
        proj[row * GAT_FIN + col] = accp[v];
        out [row * GAT_FIN + col] = accs[v] + bcol;
    }
}

// ---------------------------------------------------------------------------
// Kernel 2: per-(node, head) attention score dots; also zero denom and init
// the global-max cell (stream order guarantees visibility before kernel 3a).
// ---------------------------------------------------------------------------
__global__ __launch_bounds__(256) void gat_node_score_kernel(
    const float* __restrict__ proj,
    const float* __restrict__ w_src, const float* __restrict__ w_tgt,
    float* __restrict__ s_src, float* __restrict__ s_tgt,
    float* __restrict__ denom, float* __restrict__ gmax)
{
    const int idx = blockIdx.x * 256 + threadIdx.x;   // over N*NH
    if (idx == 0) *gmax = -INFINITY;
    if (idx >= GAT_N * GAT_NH) return;
    const int n = idx >> 3;
    const int h = idx & 7;
    const float* p  = proj  + n * GAT_FIN + h * GAT_FOUT;
    const float* wa = w_src + h * GAT_FOUT;
    const float* wb = w_tgt + h * GAT_FOUT;
    float a = 0.f, b = 0.f;
#pragma unroll
    for (int f = 0; f < GAT_FOUT; ++f) {
        a = fmaf(p[f], wa[f], a);
        b = fmaf(p[f], wb[f], b);
    }
    s_src[idx] = a;
    s_tgt[idx] = b;
    denom[idx] = 0.f;
}

__device__ __forceinline__ void gat_atomic_max_f32(float* addr, float val)
{
    unsigned int* ia  = (unsigned int*)addr;
    unsigned int  old = *ia;
    while (true) {
        if (__uint_as_float(old) >= val) break;
        unsigned int assumed = old;
        old = atomicCAS(ia, assumed, __float_as_uint(val));
        if (old == assumed) break;
    }
}

// ---------------------------------------------------------------------------
// Kernel 3a: raw edge scores (leaky relu) + global max reduction.
// ---------------------------------------------------------------------------
__global__ __launch_bounds__(256) void gat_edge_score_kernel(
    const long long* __restrict__ edges,
    const float* __restrict__ s_src, const float* __restrict__ s_tgt,
    float* __restrict__ scores, float* __restrict__ gmax)
{
    const int idx = blockIdx.x * 256 + threadIdx.x;   // over E*NH
    float s = -INFINITY;
    if (idx < GAT_E * GAT_NH) {
        const int e = idx >> 3;
        const int h = idx & 7;
        const int src = (int)edges[e];
        const int tgt = (int)edges[GAT_E + e];
        const float v = s_src[src * GAT_NH + h] + s_tgt[tgt * GAT_NH + h];
        s = (v > 0.f) ? v : v * GAT_NEG_SLOPE;
        scores[idx] = s;
    }
    // wave32 max reduction, one atomic per wave
#pragma unroll
    for (int off = 16; off > 0; off >>= 1)
        s = fmaxf(s, __shfl_xor(s, off, 32));
    if ((threadIdx.x & 31) == 0) gat_atomic_max_f32(gmax, s);
}

// ---------------------------------------------------------------------------
// Kernel 3b: exp(score - gmax) in place + segment-sum denominator.
// ---------------------------------------------------------------------------
__global__ __launch_bounds__(256) void gat_exp_denom_kernel(
    const long long* __restrict__ edges, float* __restrict__ scores,
    const float* __restrict__ gmax, float* __restrict__ denom)
{
    const int idx = blockIdx.x * 256 + threadIdx.x;   // over E*NH
    if (idx >= GAT_E * GAT_NH) return;
    const int e = idx >> 3;
    const int h = idx & 7;
    const int tgt = (int)edges[GAT_E + e];
    const float es = __expf(scores[idx] - *gmax);
    scores[idx] = es;
    atomicAdd(&denom[tgt * GAT_NH + h], es);
}

// ---------------------------------------------------------------------------
// Kernel 4: weighted gather/scatter aggregation.
// Thread (e, q) handles 4 consecutive features c = 4q of edge e
// (q in [0,32), head h = q/4): out[tgt, c..c+3] += proj[src, c..c+3] * att.
// ---------------------------------------------------------------------------
__global__ __launch_bounds__(256) void gat_aggregate_kernel(
    const long long* __restrict__ edges, const float* __restrict__ proj,
    const float* __restrict__ scores, const float* __restrict__ denom,
    float* __restrict__ out)
{
    const int gid = blockIdx.x * 256 + threadIdx.x;   // over E*32
    const int e = gid >> 5;
    const int q = gid & 31;
    const int c = q << 2;
    const int h = q >> 2;
    const int src = (int)edges[e];
    const int tgt = (int)edges[GAT_E + e];
    const float att = scores[e * GAT_NH + h] / (denom[tgt * GAT_NH + h] + GAT_EPS);
    const float4 p = *(const float4*)(proj + src * GAT_FIN + c);
    float* o = out + tgt * GAT_FIN + c;
    atomicAdd(o + 0, p.x * att);
    atomicAdd(o + 1, p.y * att);
    atomicAdd(o + 2, p.z * att);
    atomicAdd(o + 3, p.w * att);
}

// ---------------------------------------------------------------------------
extern "C" void kernel_launch(void* const* d_in, const int* in_sizes, int n_in,
                              void* d_out, int out_size, void* d_ws, size_t ws_size,
                              hipStream_t stream)
{
    (void)in_sizes; (void)n_in; (void)out_size; (void)ws_size;

    const float*     x      = (const float*)d_in[0];
    const long long* edges  = (const long long*)d_in[1];  // int64 [2, E]
    const float*     W_proj = (const float*)d_in[2];
    const float*     W_skip = (const float*)d_in[3];
    const float*     w_src  = (const float*)d_in[4];      // [NH*FOUT]
    const float*     w_tgt  = (const float*)d_in[5];
    const float*     bias   = (const float*)d_in[6];
    float*           out    = (float*)d_out;

    // Workspace carve-up (floats)
    float* ws     = (float*)d_ws;
    float* proj   = ws;                                   // N*FIN
    float* scores = proj   + (size_t)GAT_N * GAT_FIN;     // E*NH
    float* s_src  = scores + (size_t)GAT_E * GAT_NH;      // N*NH
    float* s_tgt  = s_src  + (size_t)GAT_N * GAT_NH;      // N*NH
    float* denom  = s_tgt  + (size_t)GAT_N * GAT_NH;      // N*NH
    float* gmax   = denom  + (size_t)GAT_N * GAT_NH;      // 1

    // K1: WMMA dual GEMM (proj -> ws, out = skip + bias). N = 625*16.
    gat_gemm_kernel<<<GAT_N / 16, 256, 0, stream>>>(x, W_proj, W_skip, bias, proj, out);

    // K2: per-node scores + denom/gmax init.  N*NH = 80000.
    gat_node_score_kernel<<<(GAT_N * GAT_NH + 255) / 256, 256, 0, stream>>>(
        proj, w_src, w_tgt, s_src, s_tgt, denom, gmax);

    // K3a: edge scores + global max.  E*NH = 5,120,000 = 20000*256.
    gat_edge_score_kernel<<<(GAT_E * GAT_NH) / 256, 256, 0, stream>>>(
        edges, s_src, s_tgt, scores, gmax);

    // K3b: exp + denominator segment sum.
    gat_exp_denom_kernel<<<(GAT_E * GAT_NH) / 256, 256, 0, stream>>>(
        edges, scores, gmax, denom);

    // K4: weighted aggregation.  E*32 = 20,480,000 = 80000*256.
    gat_aggregate_kernel<<<(GAT_E * 32) / 256, 256, 0, stream>>>(
        edges, proj, scores, denom, out);
}